// EmbeddingLayer_16226386444785
// MI455X (gfx1250) — compile-verified
//
#include <hip/hip_runtime.h>
#include <hip/hip_bf16.h>
#include <math.h>

#define MUL       64
#define FEAT      256        // 4*MUL input row
#define OUTW      256        // 4*MUL output row
#define C_TRIP    0.57735026918962576f   // 1/sqrt(3)
#define ALPHA_C   0.08838834764831845f   // 1/sqrt(128)

// k-pair-interleaved LDS strides (DWORDs per k-pair row), chosen ≡ 32 (mod 64)
// so lanes 0-15 (khalf=0) and 16-31 (khalf=1) hit disjoint bank halves.
#define S12 288   // W1/W2: 128 cols * 2 + 32 pad ; 32 kpairs -> 9216 floats each
#define S34 160   // W3/W4:  64 cols * 2 + 32 pad ; 32 kpairs -> 5120 floats each
#define OFF_W2 9216
#define OFF_W3 18432
#define OFF_W4 23552
#define LDS_FLOATS 28672     // 112 KB

typedef __attribute__((ext_vector_type(2))) float v2f;
typedef __attribute__((ext_vector_type(8))) float v8f;

// Stage a (64 x NCOLS) row-major matrix into k-pair-interleaved LDS:
//   dst[(k>>1)*STRIDE + col*2 + (k&1)] = src[k*NCOLS + col]
template <int NCOLS, int STRIDE>
__device__ __forceinline__ void stage_swizzled(const float* __restrict__ src,
                                               float* __restrict__ dst, int tid) {
    const float4* s4 = (const float4*)src;
    const int n4 = (64 * NCOLS) >> 2;
    for (int i = tid; i < n4; i += 256) {
        float4 v = s4[i];
        int e = i << 2;
        int k = e / NCOLS;            // NCOLS is a power of two -> shift
        int col = e - k * NCOLS;
        float* p = dst + (k >> 1) * STRIDE + col * 2 + (k & 1);
        p[0] = v.x; p[2] = v.y; p[4] = v.z; p[6] = v.w;
    }
}

__global__ __launch_bounds__(256) void eqgate_wmma_kernel(
    const float* __restrict__ feats,   // (n, 256)
    const float* __restrict__ attrs,   // (n, 4)
    const float* __restrict__ W1,      // (64, 128)
    const float* __restrict__ W2,      // (64, 128)
    const float* __restrict__ W3,      // (64, 64)
    const float* __restrict__ W4,      // (64, 64)
    float* __restrict__ out,           // (n, 256)
    int nNodes)
{
    extern __shared__ float smem[];
    float* sW1 = smem;
    float* sW2 = smem + OFF_W2;
    float* sW3 = smem + OFF_W3;
    float* sW4 = smem + OFF_W4;

    const int tid = threadIdx.x;

    stage_swizzled<128, S12>(W1, sW1, tid);
    stage_swizzled<128, S12>(W2, sW2, tid);
    stage_swizzled< 64, S34>(W3, sW3, tid);
    stage_swizzled< 64, S34>(W4, sW4, tid);
    __syncthreads();

    const int wave  = tid >> 5;
    const int lane  = tid & 31;
    const int nIdx  = lane & 15;   // B/C column within 16-wide tile; A row
    const int khalf = lane >> 4;   // 0 -> K pair {0,1}, 1 -> K pair {2,3}

    const int m0 = (blockIdx.x * 8 + wave) * 16;   // first node of this wave's tile
    if (m0 >= nNodes) return;                      // wave-uniform exit (after barrier)

    int node = m0 + nIdx;
    if (node > nNodes - 1) node = nNodes - 1;      // clamp tail (loads only)

    const float* __restrict__ frow = feats + (long)node * FEAT;
    const float a0  = attrs[(long)node * 4 + 0];
    const float a1x = attrs[(long)node * 4 + 1];
    const float a1y = attrs[(long)node * 4 + 2];
    const float a1z = attrs[(long)node * 4 + 3];

    // per-lane LDS bases (constant across kt; all kt/nt reachable via imm offsets)
    const float* b1base = sW1 + nIdx * 2 + khalf * S12;
    const float* b2base = sW2 + nIdx * 2 + khalf * S12;
    const float* b3base = sW3 + nIdx * 2 + khalf * S34;
    const float* b4base = sW4 + nIdx * 2 + khalf * S34;

    // x0 pairs for this lane, all 16 K-tiles: reused by phase 1 and all of phase 2
    v2f x0all[16];
    #pragma unroll
    for (int kt = 0; kt < 16; ++kt)
        x0all[kt] = *(const v2f*)(frow + kt * 4 + khalf * 2);

    const v8f vzero = {};

    // =================== Phase 1: out0 = (a0*x0)@W1 + (C*t1)@W2 ===================
    v8f acc0[8];
    #pragma unroll
    for (int t = 0; t < 8; ++t) acc0[t] = vzero;

    #pragma unroll
    for (int kt = 0; kt < 16; ++kt) {
        const int kk = kt * 4 + khalf * 2;

        v2f fa;  fa.x = x0all[kt].x * a0;  fa.y = x0all[kt].y * a0;

        v2f u0 = *(const v2f*)(frow + 64 + 3 * kk + 0);
        v2f u1 = *(const v2f*)(frow + 64 + 3 * kk + 2);
        v2f u2 = *(const v2f*)(frow + 64 + 3 * kk + 4);
        v2f fb;
        fb.x = C_TRIP * (u0.x * a1x + u0.y * a1y + u1.x * a1z);
        fb.y = C_TRIP * (u1.y * a1x + u2.x * a1y + u2.y * a1z);

        const int kofs = kt * 2 * S12;   // compile-time constant after unroll

        #pragma unroll
        for (int nt = 0; nt < 8; ++nt) {               // sweep 1: all fa WMMAs
            v2f b1 = *(const v2f*)(b1base + kofs + nt * 32);
            acc0[nt] = __builtin_amdgcn_wmma_f32_16x16x4_f32(
                false, fa, false, b1, (short)0, acc0[nt], false, false);
        }
        #pragma unroll
        for (int nt = 0; nt < 8; ++nt) {               // sweep 2: all fb WMMAs
            v2f b2 = *(const v2f*)(b2base + kofs + nt * 32);
            acc0[nt] = __builtin_amdgcn_wmma_f32_16x16x4_f32(
                false, fb, false, b2, (short)0, acc0[nt], false, false);
        }
    }

    // silu(s) -> out[:, 0:64]; keep sigmoid(g) for gating
    v8f sig[4];
    #pragma unroll
    for (int nt = 0; nt < 4; ++nt) {
        #pragma unroll
        for (int v = 0; v < 8; ++v) {
            float sv = ALPHA_C * acc0[nt][v];
            float gv = ALPHA_C * acc0[nt + 4][v];
            float ss = 1.0f / (1.0f + __expf(-sv));
            sig[nt][v] = 1.0f / (1.0f + __expf(-gv));
            int z = m0 + v + khalf * 8;
            if (z < nNodes)
                __builtin_nontemporal_store(sv * ss,
                    out + (long)z * OUTW + nt * 16 + nIdx);
        }
    }

    // ============ Phase 2: out1_i = (C*a1_i*x0)@W3 + (C*a0*x1_i)@W4, gated ============
    const float cb = C_TRIP * a0;
    #pragma unroll
    for (int i = 0; i < 3; ++i) {
        const float cai = C_TRIP * (i == 0 ? a1x : (i == 1 ? a1y : a1z));

        v8f acc1[4];
        #pragma unroll
        for (int t = 0; t < 4; ++t) acc1[t] = vzero;

        #pragma unroll
        for (int kt = 0; kt < 16; ++kt) {
            const int kk = kt * 4 + khalf * 2;

            v2f fa;  fa.x = x0all[kt].x * cai;  fa.y = x0all[kt].y * cai;

            float w0 = frow[64 + 3 * (kk + 0) + i];
            float w1 = frow[64 + 3 * (kk + 1) + i];
            v2f fb;  fb.x = w0 * cb;  fb.y = w1 * cb;

            const int kofs = kt * 2 * S34;

            #pragma unroll
            for (int nt = 0; nt < 4; ++nt) {           // sweep 1: all fa WMMAs
                v2f b3 = *(const v2f*)(b3base + kofs + nt * 32);
                acc1[nt] = __builtin_amdgcn_wmma_f32_16x16x4_f32(
                    false, fa, false, b3, (short)0, acc1[nt], false, false);
            }
            #pragma unroll
            for (int nt = 0; nt < 4; ++nt) {           // sweep 2: all fb WMMAs
                v2f b4 = *(const v2f*)(b4base + kofs + nt * 32);
                acc1[nt] = __builtin_amdgcn_wmma_f32_16x16x4_f32(
                    false, fb, false, b4, (short)0, acc1[nt], false, false);
            }
        }

        // gated store: out[z, 64 + w*3 + i], w = nt*16 + nIdx
        #pragma unroll
        for (int nt = 0; nt < 4; ++nt) {
            #pragma unroll
            for (int v = 0; v < 8; ++v) {
                int z = m0 + v + khalf * 8;
                if (z < nNodes) {
                    int w = nt * 16 + nIdx;
                    __builtin_nontemporal_store(sig[nt][v] * (ALPHA_C * acc1[nt][v]),
                        out + (long)z * OUTW + 64 + w * 3 + i);
                }
            }
        }
    }
}

extern "C" void kernel_launch(void* const* d_in, const int* in_sizes, int n_in,
                              void* d_out, int out_size, void* d_ws, size_t ws_size,
                              hipStream_t stream) {
    const float* feats = (const float*)d_in[0];
    const float* attrs = (const float*)d_in[1];
    const float* W1    = (const float*)d_in[2];
    const float* W2    = (const float*)d_in[3];
    const float* W3    = (const float*)d_in[4];
    const float* W4    = (const float*)d_in[5];
    float* out = (float*)d_out;

    const int nNodes = in_sizes[0] / FEAT;            // 200000
    const int nodesPerBlock = 128;                    // 8 waves * 16 nodes
    const int grid = (nNodes + nodesPerBlock - 1) / nodesPerBlock;
    const size_t ldsBytes = LDS_FLOATS * sizeof(float);  // 112 KB

    eqgate_wmma_kernel<<<grid, 256, ldsBytes, stream>>>(
        feats, attrs, W1, W2, W3, W4, out, nNodes);
}